// HGT_75067438399874
// MI455X (gfx1250) — compile-verified
//
#include <hip/hip_runtime.h>
#include <math.h>

typedef __attribute__((ext_vector_type(2))) float v2f;
typedef __attribute__((ext_vector_type(8))) float v8f;

#define HID 128
#define NHEAD 4
#define DHEAD 32
#define NUMG 4096
#define NB_N (19 * NUMG)      /* 77824 bus nodes  */
#define NG_N 32768            /* gen nodes        */
#define E_BB 400000
#define E_GB 100000
#define E_BG 100000
#define MROWS (NUMG * 4)      /* 16384 MLP rows   */

// ---------------------------------------------------------------------------
// WMMA f32 GEMM:  C[M,128] = epilogue(A[M,K] @ W[K,128] + bias)
// mode 0: none | 1: relu | 2: out = a*(acc+b) + (1-a)*xold, a = sigmoid(*skip)
// One wave per 16x16 tile; 8 waves/block cover all 128 columns. M % 16 == 0.
// A-frag (16x4 f32): lanes 0-15 row M=l, VGPR0=K0,VGPR1=K1; lanes 16-31 K2/K3.
// B-frag (4x16 f32): VGPR0 = B[k0+2*half, n=l%16], VGPR1 = B[k0+2*half+1, n].
// C/D: VGPR j -> C[row0 + j + 8*half, col0 + l%16].
// ---------------------------------------------------------------------------
__global__ __launch_bounds__(256)
void hgt_gemm_k(const float* __restrict__ A, const float* __restrict__ W,
                const float* __restrict__ bias, float* C,
                int M, int K, int mode, const float* __restrict__ skip,
                const float* xold)
{
  const int lane = threadIdx.x & 31;
  const int wave = threadIdx.x >> 5;
  const int col0 = wave * 16;
  const int row0 = blockIdx.x * 16;
  const int r    = lane & 15;
  const int half = lane >> 4;
  const float* arow = A + (size_t)(row0 + r) * K;

  v8f acc = {};
  #pragma unroll 4
  for (int k0 = 0; k0 < K; k0 += 4) {
    const int ka = k0 + half * 2;
    v2f a, b;
    a.x = arow[ka];
    a.y = arow[ka + 1];
    b.x = W[(size_t)ka * HID + col0 + r];
    b.y = W[(size_t)(ka + 1) * HID + col0 + r];
    acc = __builtin_amdgcn_wmma_f32_16x16x4_f32(false, a, false, b,
                                                (short)0, acc, false, false);
  }

  float alpha = 0.f;
  if (mode == 2) alpha = 1.f / (1.f + expf(-skip[0]));
  const int colC = col0 + r;
  #pragma unroll
  for (int j = 0; j < 8; ++j) {
    const int rowC = row0 + j + 8 * half;
    float v = acc[j] + bias[colC];
    if (mode == 1)      v = fmaxf(v, 0.f);
    else if (mode == 2) v = alpha * v + (1.f - alpha) * xold[(size_t)rowC * HID + colC];
    C[(size_t)rowC * HID + colC] = v;
  }
}

// ---------------------------------------------------------------------------
// Fold per-head rel matrix into projection:  Wf[i,h*32+e] = sum_d W[i,h*32+d]*R[h,d,e]
// row index 128 == bias row.
// ---------------------------------------------------------------------------
__global__ void hgt_fuse_rel_k(const float* __restrict__ W, const float* __restrict__ b,
                               const float* __restrict__ R,
                               float* __restrict__ Wf, float* __restrict__ bf)
{
  int idx = blockIdx.x * blockDim.x + threadIdx.x;
  if (idx >= 129 * HID) return;
  int rowi = idx >> 7;
  int col  = idx & 127;
  int h = col >> 5, e = col & 31;
  const float* src = (rowi < 128) ? (W + (size_t)rowi * HID + h * DHEAD) : (b + h * DHEAD);
  const float* rp  = R + (size_t)h * DHEAD * DHEAD + e;   // R[h,d,e], stride 32 over d
  float s = 0.f;
  #pragma unroll
  for (int d = 0; d < DHEAD; ++d) s = fmaf(src[d], rp[(size_t)d * DHEAD], s);
  if (rowi < 128) Wf[(size_t)rowi * HID + col] = s; else bf[col] = s;
}

// --------------------------- segment softmax helpers -----------------------
__device__ __forceinline__ unsigned encF(float f) {
  unsigned u = __float_as_uint(f);
  return (u & 0x80000000u) ? ~u : (u | 0x80000000u);
}
__device__ __forceinline__ float decF(unsigned u) {
  unsigned b = (u & 0x80000000u) ? (u & 0x7FFFFFFFu) : ~u;
  return __uint_as_float(b);
}

__global__ void hgt_edge_logits_k(const int* __restrict__ src, const int* __restrict__ dst,
                                  int E, const float* __restrict__ q,
                                  const float* __restrict__ kr, const float* __restrict__ prel,
                                  float* __restrict__ logbuf, unsigned* __restrict__ maxbuf)
{
  int t = blockIdx.x * blockDim.x + threadIdx.x;
  if (t >= E * NHEAD) return;
  int e = t >> 2, h = t & 3;
  int s = src[e], d = dst[e];
  const float4* qp = (const float4*)(q  + (size_t)d * HID + h * DHEAD);
  const float4* kp = (const float4*)(kr + (size_t)s * HID + h * DHEAD);
  float acc = 0.f;
  #pragma unroll
  for (int i = 0; i < DHEAD / 4; ++i) {
    float4 a = qp[i], b = kp[i];
    acc = fmaf(a.x, b.x, acc); acc = fmaf(a.y, b.y, acc);
    acc = fmaf(a.z, b.z, acc); acc = fmaf(a.w, b.w, acc);
  }
  acc *= prel[h] * 0.17677669529663687f;   // 1/sqrt(32)
  logbuf[t] = acc;
  atomicMax(maxbuf + (size_t)d * NHEAD + h, encF(acc));
}

__global__ void hgt_edge_expsum_k(const int* __restrict__ dst, int E,
                                  float* __restrict__ logbuf,
                                  const unsigned* __restrict__ maxbuf,
                                  float* __restrict__ sumbuf)
{
  int t = blockIdx.x * blockDim.x + threadIdx.x;
  if (t >= E * NHEAD) return;
  int e = t >> 2, h = t & 3;
  int d = dst[e];
  float m  = decF(maxbuf[(size_t)d * NHEAD + h]);
  float el = expf(logbuf[t] - m);
  logbuf[t] = el;
  atomicAdd(sumbuf + (size_t)d * NHEAD + h, el);
}

__global__ void hgt_edge_msg_k(const int* __restrict__ src, const int* __restrict__ dst,
                               int E, const float* __restrict__ logbuf,
                               const float* __restrict__ sumbuf,
                               const float* __restrict__ vr, float* __restrict__ out)
{
  int t = blockIdx.x * blockDim.x + threadIdx.x;
  if (t >= E * NHEAD) return;
  int e = t >> 2, h = t & 3;
  int s = src[e], d = dst[e];
  float coef = logbuf[t] / (sumbuf[(size_t)d * NHEAD + h] + 1e-16f);
  const float* vp = vr + (size_t)s * HID + h * DHEAD;
  float* op = out + (size_t)d * HID + h * DHEAD;
  #pragma unroll
  for (int i = 0; i < DHEAD; ++i) atomicAdd(op + i, vp[i] * coef);
}

// ------------------------------ elementwise --------------------------------
__device__ __forceinline__ float gelu_exact(float x) {
  return 0.5f * x * (1.f + erff(x * 0.70710678118654752f));
}
__global__ void hgt_gelu_min_k(float* a, const float* __restrict__ b, int n) {
  int i = blockIdx.x * blockDim.x + threadIdx.x;
  if (i < n) a[i] = gelu_exact(fminf(a[i], b[i]));
}
__global__ void hgt_gelu_k(float* a, int n) {
  int i = blockIdx.x * blockDim.x + threadIdx.x;
  if (i < n) a[i] = gelu_exact(a[i]);
}
__global__ void hgt_gather_k(const float* __restrict__ xb, const int* __restrict__ nidx,
                             float* __restrict__ h0) {
  int i = blockIdx.x * blockDim.x + threadIdx.x;
  if (i >= MROWS * HID) return;
  int r = i >> 7, c = i & 127;
  int node = 19 * (r >> 2) + nidx[r & 3];
  h0[i] = xb[(size_t)node * HID + c];
}

// ------------------------------- batchnorm ---------------------------------
__global__ __launch_bounds__(256)
void hgt_bn_stats_k(const float* __restrict__ h, float* __restrict__ mean,
                    float* __restrict__ rstd) {
  __shared__ float sh[256], sh2[256];
  int c = blockIdx.x, tid = threadIdx.x;
  float s = 0.f, s2 = 0.f;
  for (int r = tid; r < MROWS; r += 256) {
    float v = h[(size_t)r * HID + c];
    s += v; s2 = fmaf(v, v, s2);
  }
  sh[tid] = s; sh2[tid] = s2; __syncthreads();
  for (int w = 128; w > 0; w >>= 1) {
    if (tid < w) { sh[tid] += sh[tid + w]; sh2[tid] += sh2[tid + w]; }
    __syncthreads();
  }
  if (tid == 0) {
    float mu = sh[0] / (float)MROWS;
    float var = sh2[0] / (float)MROWS - mu * mu;
    mean[c] = mu;
    rstd[c] = rsqrtf(fmaxf(var, 0.f) + 1e-5f);
  }
}
__global__ void hgt_bn_apply_k(float* h, const float* __restrict__ mean,
                               const float* __restrict__ rstd,
                               const float* __restrict__ g, const float* __restrict__ b) {
  int i = blockIdx.x * blockDim.x + threadIdx.x;
  if (i >= MROWS * HID) return;
  int c = i & 127;
  float v = g[c] * (h[i] - mean[c]) * rstd[c] + b[c];
  h[i] = fmaxf(v, 0.f);
}

// ------------------------------ final 128->4 -------------------------------
__global__ void hgt_final_k(const float* __restrict__ h, const float* __restrict__ W4,
                            const float* __restrict__ b4, float* __restrict__ out) {
  int i = blockIdx.x * blockDim.x + threadIdx.x;
  if (i >= MROWS * 4) return;
  int r = i >> 2, c = i & 3;
  const float* hr = h + (size_t)r * HID;
  float s = b4[c];
  #pragma unroll 8
  for (int k = 0; k < HID; ++k) s = fmaf(hr[k], W4[k * 4 + c], s);
  out[i] = s;
}

// ===========================================================================
extern "C" void kernel_launch(void* const* d_in, const int* in_sizes, int n_in,
                              void* d_out, int out_size, void* d_ws, size_t ws_size,
                              hipStream_t stream)
{
  const float* x_bus    = (const float*)d_in[0];
  const float* x_gen    = (const float*)d_in[1];
  const int*   ei_bb    = (const int*)d_in[2];
  const int*   ei_gb    = (const int*)d_in[3];
  const int*   ei_bg    = (const int*)d_in[4];
  const int*   node_idx = (const int*)d_in[5];
  auto P = [&](int i) { return (const float*)d_in[i]; };
  // params flat order: 6..9 input lin; conv layer L at base=10+27L; mlp at 64.

  // ---- workspace carve (floats) ----
  float* ws = (float*)d_ws;
  size_t off = 0;
  auto alloc = [&](size_t n) { float* q = ws + off; off += n; return q; };
  float*    xb     = alloc((size_t)NB_N * HID);
  float*    xg     = alloc((size_t)NG_N * HID);
  float*    q_bus  = alloc((size_t)NB_N * HID);
  float*    q_gen  = alloc((size_t)NG_N * HID);
  float*    krbuf  = alloc((size_t)NB_N * HID);   // reused for gen source too
  float*    vrbuf  = alloc((size_t)NB_N * HID);
  float*    out_bb = alloc((size_t)NB_N * HID);
  float*    out_gb = alloc((size_t)NB_N * HID);
  float*    out_bg = alloc((size_t)NG_N * HID);
  float*    Wf_k   = alloc(HID * HID);
  float*    bf_k   = alloc(HID);
  float*    Wf_v   = alloc(HID * HID);
  float*    bf_v   = alloc(HID);
  float*    logbuf = alloc((size_t)E_BB * NHEAD);
  unsigned* maxbuf = (unsigned*)alloc((size_t)NB_N * NHEAD);
  float*    sumbuf = alloc((size_t)NB_N * NHEAD);
  float*    h0     = alloc((size_t)MROWS * HID);
  float*    h1     = alloc((size_t)MROWS * HID);
  float*    meanb  = alloc(HID);
  float*    rstdb  = alloc(HID);
  (void)ws_size; (void)in_sizes; (void)n_in; (void)out_size;

  // ---- input linears (relu) ----
  hgt_gemm_k<<<NB_N / 16, 256, 0, stream>>>(x_bus, P(6), P(7), xb, NB_N, 16, 1, nullptr, nullptr);
  hgt_gemm_k<<<NG_N / 16, 256, 0, stream>>>(x_gen, P(8), P(9), xg, NG_N, 8, 1, nullptr, nullptr);

  // ---- HGT conv layers ----
  for (int L = 0; L < 2; ++L) {
    const int base = 10 + L * 27;
    hgt_gemm_k<<<NB_N / 16, 256, 0, stream>>>(xb, P(base + 4), P(base + 5), q_bus, NB_N, HID, 0, nullptr, nullptr);
    hgt_gemm_k<<<NG_N / 16, 256, 0, stream>>>(xg, P(base + 6), P(base + 7), q_gen, NG_N, HID, 0, nullptr, nullptr);
    hipMemsetAsync(out_bb, 0, (size_t)NB_N * HID * 4, stream);
    hipMemsetAsync(out_gb, 0, (size_t)NB_N * HID * 4, stream);
    hipMemsetAsync(out_bg, 0, (size_t)NG_N * HID * 4, stream);

    for (int t = 0; t < 3; ++t) {   // 0: bb (bus->bus), 1: gb (gen->bus), 2: bg (bus->gen)
      const int*   ei;  int E, nsrc, ndst, kW, kB, vW, vB;
      const float* xsrc; const float* qd; float* oac;
      if (t == 0) { ei = ei_bb; E = E_BB; xsrc = xb; qd = q_bus; nsrc = NB_N; ndst = NB_N;
                    oac = out_bb; kW = base + 0; kB = base + 1; vW = base + 8;  vB = base + 9; }
      else if (t == 1) { ei = ei_gb; E = E_GB; xsrc = xg; qd = q_bus; nsrc = NG_N; ndst = NB_N;
                    oac = out_gb; kW = base + 2; kB = base + 3; vW = base + 10; vB = base + 11; }
      else        { ei = ei_bg; E = E_BG; xsrc = xb; qd = q_gen; nsrc = NB_N; ndst = NG_N;
                    oac = out_bg; kW = base + 0; kB = base + 1; vW = base + 8;  vB = base + 9; }
      const float* arel = P(base + 18 + t);
      const float* mrel = P(base + 21 + t);
      const float* prel = P(base + 24 + t);

      hgt_fuse_rel_k<<<(129 * HID + 255) / 256, 256, 0, stream>>>(P(kW), P(kB), arel, Wf_k, bf_k);
      hgt_fuse_rel_k<<<(129 * HID + 255) / 256, 256, 0, stream>>>(P(vW), P(vB), mrel, Wf_v, bf_v);
      hgt_gemm_k<<<nsrc / 16, 256, 0, stream>>>(xsrc, Wf_k, bf_k, krbuf, nsrc, HID, 0, nullptr, nullptr);
      hgt_gemm_k<<<nsrc / 16, 256, 0, stream>>>(xsrc, Wf_v, bf_v, vrbuf, nsrc, HID, 0, nullptr, nullptr);

      hipMemsetAsync(maxbuf, 0, (size_t)ndst * NHEAD * 4, stream);
      hipMemsetAsync(sumbuf, 0, (size_t)ndst * NHEAD * 4, stream);
      const int nt = E * NHEAD, nb = (nt + 255) / 256;
      hgt_edge_logits_k<<<nb, 256, 0, stream>>>(ei, ei + E, E, qd, krbuf, prel, logbuf, maxbuf);
      hgt_edge_expsum_k<<<nb, 256, 0, stream>>>(ei + E, E, logbuf, maxbuf, sumbuf);
      hgt_edge_msg_k<<<nb, 256, 0, stream>>>(ei, ei + E, E, logbuf, sumbuf, vrbuf, oac);
    }

    // combine (group='min' for bus), gelu, output proj + sigmoid-skip
    hgt_gelu_min_k<<<((size_t)NB_N * HID + 255) / 256, 256, 0, stream>>>(out_bb, out_gb, NB_N * HID);
    hgt_gelu_k<<<((size_t)NG_N * HID + 255) / 256, 256, 0, stream>>>(out_bg, NG_N * HID);
    hgt_gemm_k<<<NB_N / 16, 256, 0, stream>>>(out_bb, P(base + 12), P(base + 13), xb, NB_N, HID, 2, P(base + 16), xb);
    hgt_gemm_k<<<NG_N / 16, 256, 0, stream>>>(out_bg, P(base + 14), P(base + 15), xg, NG_N, HID, 2, P(base + 17), xg);
  }

  // ---- gather + MLP with BatchNorm ----
  hgt_gather_k<<<(MROWS * HID + 255) / 256, 256, 0, stream>>>(xb, node_idx, h0);
  float* hc = h0; float* hn = h1;
  for (int i = 0; i < 4; ++i) {
    hgt_gemm_k<<<MROWS / 16, 256, 0, stream>>>(hc, P(64 + i), P(69 + i), hn, MROWS, HID, 0, nullptr, nullptr);
    hgt_bn_stats_k<<<HID, 256, 0, stream>>>(hn, meanb, rstdb);
    hgt_bn_apply_k<<<(MROWS * HID + 255) / 256, 256, 0, stream>>>(hn, meanb, rstdb, P(74 + i), P(78 + i));
    float* tmp = hc; hc = hn; hn = tmp;
  }
  hgt_final_k<<<(MROWS * 4 + 255) / 256, 256, 0, stream>>>(hc, P(68), P(73), (float*)d_out);
}